// QueryAndGroupV2_5892695130409
// MI455X (gfx1250) — compile-verified
//
#include <hip/hip_runtime.h>

#define B_ 8
#define N_ 16384
#define S_ 2048
#define C_ 64
#define K_ 32                    // SAMPLE_NUM
#define R2_ 0.04f                // MAX_RADIUS^2
#define WPB 8                    // waves per block (wave32)
#define THREADS (WPB * 32)
#define CHUNK 2048               // points per staged LDS tile
#define CHUNK_BYTES (CHUNK * 12) // 24576 B (xyz fp32)
#define NCHUNKS (N_ / CHUNK)     // 8

// Stage CHUNK_BYTES from global into LDS using CDNA5 async copies.
// 256 threads x 16B = 4 KB per issue -> 6 issues per tile. Tracked on ASYNCcnt.
__device__ __forceinline__ void stage_chunk(const char* gsrc, float* ldst, int tid) {
#pragma unroll
    for (int u = 0; u < CHUNK_BYTES / (THREADS * 16); ++u) {
        const int off = u * (THREADS * 16) + tid * 16;
        unsigned lds_addr = (unsigned)(size_t)(const void*)((const char*)ldst + off);
        const char* g = gsrc + off;
        asm volatile("global_load_async_to_lds_b128 %0, %1, off"
                     :: "v"(lds_addr), "v"(g) : "memory");
    }
}

__device__ __forceinline__ void wait_async0() {
    asm volatile("s_wait_asynccnt 0x0" ::: "memory");
}

__global__ __launch_bounds__(THREADS)
void qg_ballquery_group_kernel(const float* __restrict__ pts,   // (B,N,3)
                               const float* __restrict__ ctr,   // (B,S,3)
                               const float* __restrict__ feat,  // (B,C,N)
                               float* __restrict__ out)         // (B,67,S,32)
{
    __shared__ float buf[2][CHUNK * 3];     // 48 KB double-buffered point tiles
    __shared__ int   selList[WPB][K_];
    __shared__ int   doneCnt;

    const int tid  = threadIdx.x;
    const int wave = tid >> 5;
    const int lane = tid & 31;

    // 8 consecutive centers of one batch per block (S % WPB == 0)
    const int b = blockIdx.x / (S_ / WPB);
    const int s = (blockIdx.x % (S_ / WPB)) * WPB + wave;

    if (tid == 0) doneCnt = 0;

    const size_t cOff = ((size_t)b * S_ + s) * 3;
    const float cx = ctr[cOff + 0];
    const float cy = ctr[cOff + 1];
    const float cz = ctr[cOff + 2];

    const char* pbase = (const char*)pts + (size_t)b * N_ * 3 * sizeof(float);

    // Prefetch tile 0, then pipeline: stage tile t+1 while scanning tile t.
    stage_chunk(pbase, &buf[0][0], tid);
    wait_async0();
    __syncthreads();

    int  cnt    = 0;
    bool myDone = false;

    for (int chunk = 0; chunk < NCHUNKS; ++chunk) {
        const int cur = chunk & 1;
        if (chunk + 1 < NCHUNKS)
            stage_chunk(pbase + (size_t)(chunk + 1) * CHUNK_BYTES, &buf[cur ^ 1][0], tid);

        if (!myDone) {
            const float* pb = &buf[cur][0];
            for (int g = 0; g < CHUNK / 32; ++g) {
                const int j = g * 32 + lane;           // lane order == index order
                const float dx = pb[3 * j + 0] - cx;
                const float dy = pb[3 * j + 1] - cy;
                const float dz = pb[3 * j + 2] - cz;
                const float d2 = dx * dx + dy * dy + dz * dz;
                const bool hit = d2 < R2_;
                const unsigned m = __builtin_amdgcn_ballot_w32(hit);
                const int pos = cnt + __popc(m & ((1u << lane) - 1u));
                if (hit && pos < K_)
                    selList[wave][pos] = chunk * CHUNK + j;  // ascending index order
                cnt += __popc(m);
                if (cnt >= K_) break;                   // early exit: first 32 found
            }
            if (cnt >= K_) {
                myDone = true;
                if (lane == 0) atomicAdd(&doneCnt, 1);
            }
        }

        wait_async0();       // my async issues done before the barrier...
        __syncthreads();     // ...so after it, everyone's tile data is in LDS
        if (doneCnt >= WPB) break;   // whole block finished early
    }

    // Reference fill semantics: slots >= cnt -> first hit; no hits at all -> 0.
    int sel;
    if (cnt == 0) sel = 0;
    else          sel = selList[wave][(lane < cnt) ? lane : 0];

    // ---- gather + write: out[b, ch, s, k], stores coalesced per channel ----
    const size_t chStride = (size_t)S_ * K_;
    size_t o = (((size_t)b * 67) * S_ + s) * K_ + lane;

    const float* pB = pts + (size_t)b * N_ * 3;
    const float px = pB[3 * sel + 0];
    const float py = pB[3 * sel + 1];
    const float pz = pB[3 * sel + 2];
    __builtin_nontemporal_store(px - cx, &out[o]);  o += chStride;
    __builtin_nontemporal_store(py - cy, &out[o]);  o += chStride;
    __builtin_nontemporal_store(pz - cz, &out[o]);  o += chStride;

    const float* fB = feat + (size_t)b * C_ * N_;
#pragma unroll 8
    for (int ch = 0; ch < C_; ++ch) {
        __builtin_nontemporal_store(fB[(size_t)ch * N_ + sel], &out[o]);
        o += chStride;
    }
}

extern "C" void kernel_launch(void* const* d_in, const int* in_sizes, int n_in,
                              void* d_out, int out_size, void* d_ws, size_t ws_size,
                              hipStream_t stream) {
    (void)in_sizes; (void)n_in; (void)out_size; (void)d_ws; (void)ws_size;
    const float* pts  = (const float*)d_in[0];   // points_xyz (8,16384,3)
    const float* ctr  = (const float*)d_in[1];   // center_xyz (8,2048,3)
    const float* feat = (const float*)d_in[2];   // features   (8,64,16384)
    float* out = (float*)d_out;                  // (8,67,2048,32)

    dim3 grid(B_ * S_ / WPB);                    // 2048 blocks
    qg_ballquery_group_kernel<<<grid, THREADS, 0, stream>>>(pts, ctr, feat, out);
}